// BSplineField3d_42391327212040
// MI455X (gfx1250) — compile-verified
//
#include <hip/hip_runtime.h>
#include <hip/hip_bf16.h>

// 3D cubic B-spline field evaluation, MI455X (gfx1250, wave32).
//
// Roofline: control grid (25 MB) is fully L2-resident (192 MB L2); the 2M x 64
// random gathers dominate -> optimize the gather pipe, not FLOPs.
//  - Pre-pass pads control grid [128^3,3] -> [128^3,4]: each tap is ONE
//    global_load_b128 instead of 3 dword gathers (3x fewer gather ops).
//  - Point coordinates double-buffered through LDS with CDNA5 async
//    global->LDS loads (ASYNCcnt + s_wait_asynccnt), overlapped with compute.
//  - Output written as a single nontemporal b96 store per point (streamed
//    once; NT keeps the L2 reserved for the resident grid).

#define GRID_N 128
#define NCOMP  3
#define BLOCK  256

typedef float vfloat3 __attribute__((ext_vector_type(3)));

__device__ __forceinline__ void bspline_w(float u, float w[4]) {
    const float u2 = u * u;
    const float u3 = u2 * u;
    const float om = 1.0f - u;
    w[0] = om * om * om * (1.0f / 6.0f);
    w[1] = fmaf(3.0f, u3, fmaf(-6.0f, u2, 4.0f)) * (1.0f / 6.0f);
    w[2] = fmaf(-3.0f, u3, fmaf(3.0f, u2, fmaf(3.0f, u, 1.0f))) * (1.0f / 6.0f);
    w[3] = u3 * (1.0f / 6.0f);
}

__device__ __forceinline__ int clamp_i(int v, int lo, int hi) {
    return v < lo ? lo : (v > hi ? hi : v);
}

// Repack phi [ncell,3] -> [ncell,4] so each tap is one aligned 16B load.
__global__ void pad_phi_kernel(const float* __restrict__ phi,
                               float4* __restrict__ phi4, int ncell) {
    int i = blockIdx.x * blockDim.x + threadIdx.x;
    if (i < ncell) {
        float4 v;
        v.x = phi[3 * i + 0];
        v.y = phi[3 * i + 1];
        v.z = phi[3 * i + 2];
        v.w = 0.0f;
        phi4[i] = v;
    }
}

template <bool PADDED>
__global__ __launch_bounds__(BLOCK) void spline_eval_kernel(
    const float* __restrict__ xs, const float* __restrict__ ys,
    const float* __restrict__ zs, const float* __restrict__ phi,
    const float4* __restrict__ phi4, float* __restrict__ out,
    int n, int nChunks) {
    // Double-buffered per-lane coordinate staging: [buf][axis][tid].
    __shared__ float sbuf[2][3][BLOCK];
    const int tid = threadIdx.x;

    // Issue CDNA5 async global->LDS loads (tracked by ASYNCcnt) for one chunk.
    // Each lane stages only its own slot, so no cross-wave/LDS barrier needed;
    // ordering is handled with counted s_wait_asynccnt (in-order completion).
    auto stage = [&](int chunk, int buf) {
        int p = chunk * BLOCK + tid;
        p = p < (n - 1) ? p : (n - 1);  // tail lanes re-load a valid address
        unsigned l0 = (unsigned)(size_t)&sbuf[buf][0][tid];
        unsigned l1 = (unsigned)(size_t)&sbuf[buf][1][tid];
        unsigned l2 = (unsigned)(size_t)&sbuf[buf][2][tid];
        unsigned long long ax = (unsigned long long)(size_t)(xs + p);
        unsigned long long ay = (unsigned long long)(size_t)(ys + p);
        unsigned long long az = (unsigned long long)(size_t)(zs + p);
        asm volatile("global_load_async_to_lds_b32 %0, %1, off"
                     :: "v"(l0), "v"(ax) : "memory");
        asm volatile("global_load_async_to_lds_b32 %0, %1, off"
                     :: "v"(l1), "v"(ay) : "memory");
        asm volatile("global_load_async_to_lds_b32 %0, %1, off"
                     :: "v"(l2), "v"(az) : "memory");
    };

    int chunk = blockIdx.x;
    if (chunk >= nChunks) return;

    stage(chunk, 0);
    int cur = 0;

    for (; chunk < nChunks; chunk += gridDim.x) {
        const int nxt = chunk + (int)gridDim.x;
        if (nxt < nChunks) {
            stage(nxt, cur ^ 1);
            // 3 newest asyncs belong to the NEXT buffer; <=3 outstanding
            // guarantees the 3 for the CURRENT buffer have landed in LDS.
            asm volatile("s_wait_asynccnt 0x3" ::: "memory");
        } else {
            asm volatile("s_wait_asynccnt 0x0" ::: "memory");
        }

        const float x = sbuf[cur][0][tid];
        const float y = sbuf[cur][1][tid];
        const float z = sbuf[cur][2][tid];

        // u = (x + 1) * (N-3)/2 ; spacing 2/(N-3), origin -1-dx (reference).
        const float scale = (float)(GRID_N - 3) * 0.5f;  // 62.5 for N=128
        float u = fmaf(x, scale, scale);
        float v = fmaf(y, scale, scale);
        float w = fmaf(z, scale, scale);
        const float fu = floorf(u), fv = floorf(v), fw = floorf(w);
        const int ix = (int)fu, iy = (int)fv, iz = (int)fw;
        u -= fu; v -= fv; w -= fw;

        float wu[4], wv[4], ww[4];
        bspline_w(u, wu);
        bspline_w(v, wv);
        bspline_w(w, ww);

        float accx = 0.0f, accy = 0.0f, accz = 0.0f;

#pragma unroll
        for (int l = 0; l < 4; ++l) {
            const int xi = clamp_i(ix + l, 0, GRID_N - 1);
            const int xoff = xi * GRID_N;
#pragma unroll
            for (int m = 0; m < 4; ++m) {
                const int yi = clamp_i(iy + m, 0, GRID_N - 1);
                const float wlm = wu[l] * wv[m];
                const int rowbase = (xoff + yi) * GRID_N;
#pragma unroll
                for (int nn = 0; nn < 4; ++nn) {
                    const int zi = clamp_i(iz + nn, 0, GRID_N - 1);
                    const float c = wlm * ww[nn];
                    if (PADDED) {
                        const float4 g = phi4[rowbase + zi];  // one b128 gather
                        accx = fmaf(c, g.x, accx);
                        accy = fmaf(c, g.y, accy);
                        accz = fmaf(c, g.z, accz);
                    } else {
                        const int b = (rowbase + zi) * 3;
                        accx = fmaf(c, phi[b + 0], accx);
                        accy = fmaf(c, phi[b + 1], accy);
                        accz = fmaf(c, phi[b + 2], accz);
                    }
                }
            }
        }

        const int p = chunk * BLOCK + tid;
        if (p < n) {
            // One nontemporal 12-byte store per point -> global_store_b96.
            vfloat3 r;
            r.x = accx; r.y = accy; r.z = accz;
            __builtin_nontemporal_store(r, (vfloat3*)(out + 3 * (size_t)p));
        }
        cur ^= 1;
    }
}

extern "C" void kernel_launch(void* const* d_in, const int* in_sizes, int n_in,
                              void* d_out, int out_size, void* d_ws, size_t ws_size,
                              hipStream_t stream) {
    const float* xs  = (const float*)d_in[0];
    const float* ys  = (const float*)d_in[1];
    const float* zs  = (const float*)d_in[2];
    const float* phi = (const float*)d_in[3];
    float* out = (float*)d_out;

    const int n = in_sizes[0];
    const int ncell = GRID_N * GRID_N * GRID_N;
    const int nChunks = (n + BLOCK - 1) / BLOCK;
    // Enough blocks to fill the chip; each block iterates ~4 chunks so the
    // async prefetch of chunk i+1 overlaps the gather work of chunk i.
    const int blocks = nChunks < 2048 ? nChunks : 2048;

    const bool use_pad = ws_size >= (size_t)ncell * sizeof(float4);
    if (use_pad) {
        pad_phi_kernel<<<(ncell + 255) / 256, 256, 0, stream>>>(
            phi, (float4*)d_ws, ncell);
        spline_eval_kernel<true><<<blocks, BLOCK, 0, stream>>>(
            xs, ys, zs, phi, (const float4*)d_ws, out, n, nChunks);
    } else {
        spline_eval_kernel<false><<<blocks, BLOCK, 0, stream>>>(
            xs, ys, zs, phi, (const float4*)nullptr, out, n, nChunks);
    }
}